// MSQCR_89215060672689
// MI455X (gfx1250) — compile-verified
//
#include <hip/hip_runtime.h>
#include <math.h>

typedef __attribute__((ext_vector_type(2))) float v2f;
typedef __attribute__((ext_vector_type(8))) float v8f;

// ---------------------------------------------------------------------------
// Fused conv1d(1->8) + bias + ReLU + (optional maxpool2) + linear(->2) branch.
// One wave = one (row, segment).  Conv is an implicit GEMM on
// V_WMMA_F32_16X16X4_F32: M=16 positions, N=16 channels (8 used, rest exactly
// zero), K=taps zero-padded to 4 per chunk.  A-fragments are software-
// pipelined out of LDS so the ds_load wait hides under the epilogue; the
// epilogue is branchless (lanes n>=8 produce exact zeros).
// ---------------------------------------------------------------------------
template<int K, int L, int POOL, int SEG>
__global__ __launch_bounds__(256) void fused_branch_kernel(
    const float* __restrict__ x,      // [B, L]
    const float* __restrict__ convw,  // [8, 1, K]
    const float* __restrict__ convb,  // [8]
    const float* __restrict__ linw,   // [2, 8*(L/POOL)]
    float* __restrict__ part)         // [B*SEGS, 2] per-wave partial sums
{
    constexpr int PAD    = (K - 1) / 2;
    constexpr int CHUNKS = (K + 3) / 4;       // 1 for k=3, 2 for k=5
    constexpr int SEGS   = L / SEG;
    constexpr int LBUF   = SEG + 16;          // 8-elem halo each side
    constexpr int LP     = L / POOL;          // per-channel flattened length
    constexpr int NT     = SEG / 16;          // 16-position tiles per segment

    __shared__ float smem[8 * LBUF];

    const int lane  = threadIdx.x & 31;
    const int wslot = threadIdx.x >> 5;
    const int wave  = blockIdx.x * 8 + wslot;
    const int row   = wave / SEGS;
    const int seg   = wave % SEGS;
    const int segStart = seg * SEG;

    // ---- stage input segment (+halo, zero padded) into LDS -----------------
    float* buf = &smem[wslot * LBUF];
    const float* xrow = x + (size_t)row * L;
    for (int i = lane; i < LBUF; i += 32) {
        int p = segStart - 8 + i;
        buf[i] = (p >= 0 && p < L) ? xrow[p] : 0.0f;
    }
    __syncthreads();

    const int n  = lane & 15;   // WMMA N (channel) / A-matrix M (position)
    const int hi = lane >> 4;   // lane half selects K / M sub-range
    const int nc = n & 7;       // clamped channel for weight addressing

    // ---- B matrix: conv weights, taps zero-padded to 4 per chunk -----------
    // lane holds column n; VGPR v holds K = 2*hi + v (+4 for chunk 1).
    // Columns n>=8 are all-zero -> their accumulators are exactly 0.
    v2f B0, B1;
    {
        int k0 = 2 * hi, k1 = 2 * hi + 1;
        B0.x = (n < 8 && k0     < K) ? convw[n * K + k0]     : 0.0f;
        B0.y = (n < 8 && k1     < K) ? convw[n * K + k1]     : 0.0f;
        B1.x = (n < 8 && k0 + 4 < K) ? convw[n * K + k0 + 4] : 0.0f;
        B1.y = (n < 8 && k1 + 4 < K) ? convw[n * K + k1 + 4] : 0.0f;
    }
    const float bias = (n < 8) ? convb[n] : 0.0f;

    const float4* lw4 = reinterpret_cast<const float4*>(linw);

    // ---- software pipeline: prefetch A fragments for tile 0 ----------------
    // A: lane holds row m=n, VGPR v holds tap k = 2*hi + v (+4 chunk 1)
    const int abase0 = 8 + n + 2 * hi - PAD;
    v2f A0, A1;
    A0.x = buf[abase0];     A0.y = buf[abase0 + 1];
    if (CHUNKS > 1) { A1.x = buf[abase0 + 4]; A1.y = buf[abase0 + 5]; }

    float lacc0 = 0.0f, lacc1 = 0.0f;

    for (int t = 0; t < NT; ++t) {
        // prefetch next tile's A fragments (clamped on last iter; unused)
        const int tn  = (t + 1 < NT) ? (t + 1) : t;
        const int abn = abase0 + 16 * tn;
        v2f nA0, nA1;
        nA0.x = buf[abn];     nA0.y = buf[abn + 1];
        if (CHUNKS > 1) { nA1.x = buf[abn + 4]; nA1.y = buf[abn + 5]; }

        v8f acc;
        #pragma unroll
        for (int i = 0; i < 8; ++i) acc[i] = bias;   // C = conv bias

        acc = __builtin_amdgcn_wmma_f32_16x16x4_f32(
                  false, A0, false, B0, (short)0, acc, false, false);
        if (CHUNKS > 1)
            acc = __builtin_amdgcn_wmma_f32_16x16x4_f32(
                      false, A1, false, B1, (short)0, acc, false, false);

        // D layout: lane (n + 16*hi) VGPR v = conv[ch=n, pos = p0 + 8*hi + v]
        // Branchless epilogue: lanes n>=8 hold acc==0 -> contribute exact 0.
        const int p0 = 16 * t;
        if (POOL == 2) {
            float pv[4];
            #pragma unroll
            for (int i = 0; i < 4; ++i)     // relu then maxpool2 == max(a,b,0)
                pv[i] = fmaxf(fmaxf(acc[2 * i], acc[2 * i + 1]), 0.0f);
            const int pbase = nc * LP + (segStart + p0) / 2 + 4 * hi;  // %4==0
            float4 w0 = lw4[(0 * 8 * LP + pbase) >> 2];
            float4 w1 = lw4[(1 * 8 * LP + pbase) >> 2];
            lacc0 += pv[0]*w0.x + pv[1]*w0.y + pv[2]*w0.z + pv[3]*w0.w;
            lacc1 += pv[0]*w1.x + pv[1]*w1.y + pv[2]*w1.z + pv[3]*w1.w;
        } else {
            float rv[8];
            #pragma unroll
            for (int i = 0; i < 8; ++i) rv[i] = fmaxf(acc[i], 0.0f);
            const int pbase = nc * LP + segStart + p0 + 8 * hi;        // %8==0
            float4 w0a = lw4[((0 * 8 * LP + pbase) >> 2) + 0];
            float4 w0b = lw4[((0 * 8 * LP + pbase) >> 2) + 1];
            float4 w1a = lw4[((1 * 8 * LP + pbase) >> 2) + 0];
            float4 w1b = lw4[((1 * 8 * LP + pbase) >> 2) + 1];
            lacc0 += rv[0]*w0a.x + rv[1]*w0a.y + rv[2]*w0a.z + rv[3]*w0a.w
                   + rv[4]*w0b.x + rv[5]*w0b.y + rv[6]*w0b.z + rv[7]*w0b.w;
            lacc1 += rv[0]*w1a.x + rv[1]*w1a.y + rv[2]*w1a.z + rv[3]*w1a.w
                   + rv[4]*w1b.x + rv[5]*w1b.y + rv[6]*w1b.z + rv[7]*w1b.w;
        }

        A0 = nA0;
        if (CHUNKS > 1) A1 = nA1;
    }

    // ---- wave-wide reduction of the 2 partial linear outputs ---------------
    #pragma unroll
    for (int off = 16; off > 0; off >>= 1) {
        lacc0 += __shfl_xor(lacc0, off, 32);
        lacc1 += __shfl_xor(lacc1, off, 32);
    }
    if (lane == 0) {
        part[(row * SEGS + seg) * 2 + 0] = lacc0;
        part[(row * SEGS + seg) * 2 + 1] = lacc1;
    }
}

// ---------------------------------------------------------------------------
// Sum per-segment partials + biases, simulate the 4-qubit circuit entirely in
// registers (16 complex amplitudes), compute <Z_i> and the final FC.
// State index bit for wire w is (8 >> w)  (axis 0 = MSB, matching the (2,2,2,2)
// tensor layout of the reference).
// ---------------------------------------------------------------------------
__global__ void quantum_fc_kernel(
    const float* __restrict__ shortPart,  // [B*2, 2]
    const float* __restrict__ longPart,   // [B*8, 2]
    const float* __restrict__ cs_lin_b, const float* __restrict__ cl_lin_b,
    const float* __restrict__ rw,         // [3,4,3]
    const float* __restrict__ fc_w, const float* __restrict__ fc_b,
    float* __restrict__ out, int B)
{
    int b = blockIdx.x * blockDim.x + threadIdx.x;
    if (b >= B) return;

    float f[4];
    f[0] = cs_lin_b[0]; f[1] = cs_lin_b[1];
    #pragma unroll
    for (int s = 0; s < 2; ++s) {
        f[0] += shortPart[(b * 2 + s) * 2 + 0];
        f[1] += shortPart[(b * 2 + s) * 2 + 1];
    }
    f[2] = cl_lin_b[0]; f[3] = cl_lin_b[1];
    #pragma unroll
    for (int s = 0; s < 8; ++s) {
        f[2] += longPart[(b * 8 + s) * 2 + 0];
        f[3] += longPart[(b * 8 + s) * 2 + 1];
    }

    float sr[16], si[16];
    #pragma unroll
    for (int i = 0; i < 16; ++i) { sr[i] = 0.0f; si[i] = 0.0f; }
    sr[0] = 1.0f;

    // ---- RY(pi * f[w]) on each wire (real rotation) ------------------------
    #pragma unroll
    for (int w = 0; w < 4; ++w) {
        float t = 0.5f * 3.14159265358979323846f * f[w];
        float c = cosf(t), s = sinf(t);
        const int st = 8 >> w;
        #pragma unroll
        for (int i = 0; i < 16; ++i) {
            if (!(i & st)) {
                int j = i | st;
                float a0r = sr[i], a0i = si[i], a1r = sr[j], a1i = si[j];
                sr[i] = c * a0r - s * a1r;  si[i] = c * a0i - s * a1i;
                sr[j] = s * a0r + c * a1r;  si[j] = s * a0i + c * a1i;
            }
        }
    }

    // ---- 3 StronglyEntanglingLayers: Rot per qubit + CNOT ring -------------
    #pragma unroll
    for (int l = 0; l < 3; ++l) {
        #pragma unroll
        for (int q = 0; q < 4; ++q) {
            float phi = rw[(l * 4 + q) * 3 + 0];
            float th  = rw[(l * 4 + q) * 3 + 1];
            float om  = rw[(l * 4 + q) * 3 + 2];
            float ch = cosf(0.5f * th),        sh = sinf(0.5f * th);
            float cp = cosf(0.5f * (phi + om)), sp = sinf(0.5f * (phi + om));
            float cm = cosf(0.5f * (phi - om)), sm = sinf(0.5f * (phi - om));
            // Rot = RZ(om) RY(th) RZ(phi)
            float u00r =  ch * cp, u00i = -ch * sp;
            float u01r = -sh * cm, u01i = -sh * sm;
            float u10r =  sh * cm, u10i = -sh * sm;
            float u11r =  ch * cp, u11i =  ch * sp;
            const int st = 8 >> q;
            #pragma unroll
            for (int i = 0; i < 16; ++i) {
                if (!(i & st)) {
                    int j = i | st;
                    float a0r = sr[i], a0i = si[i], a1r = sr[j], a1i = si[j];
                    sr[i] = u00r*a0r - u00i*a0i + u01r*a1r - u01i*a1i;
                    si[i] = u00r*a0i + u00i*a0r + u01r*a1i + u01i*a1r;
                    sr[j] = u10r*a0r - u10i*a0i + u11r*a1r - u11i*a1i;
                    si[j] = u10r*a0i + u10i*a0r + u11r*a1i + u11i*a1r;
                }
            }
        }
        const int r = (l % 3) + 1;
        #pragma unroll
        for (int q = 0; q < 4; ++q) {
            const int cm_ = 8 >> q;
            const int tm_ = 8 >> ((q + r) & 3);
            #pragma unroll
            for (int i = 0; i < 16; ++i) {
                if ((i & cm_) && !(i & tm_)) {
                    int j = i | tm_;
                    float tr = sr[i]; sr[i] = sr[j]; sr[j] = tr;
                    float ti = si[i]; si[i] = si[j]; si[j] = ti;
                }
            }
        }
    }

    // ---- <Z_w> and final FC -------------------------------------------------
    float z[4] = {0.0f, 0.0f, 0.0f, 0.0f};
    #pragma unroll
    for (int i = 0; i < 16; ++i) {
        float p = sr[i] * sr[i] + si[i] * si[i];
        #pragma unroll
        for (int w = 0; w < 4; ++w)
            z[w] += (i & (8 >> w)) ? -p : p;
    }
    float y = fc_b[0];
    #pragma unroll
    for (int w = 0; w < 4; ++w) y += z[w] * fc_w[w];
    out[b] = y;
}

extern "C" void kernel_launch(void* const* d_in, const int* in_sizes, int n_in,
                              void* d_out, int out_size, void* d_ws, size_t ws_size,
                              hipStream_t stream) {
    const float* xs        = (const float*)d_in[0];
    const float* xl        = (const float*)d_in[1];
    const float* cs_conv_w = (const float*)d_in[2];
    const float* cs_conv_b = (const float*)d_in[3];
    const float* cs_lin_w  = (const float*)d_in[4];
    const float* cs_lin_b  = (const float*)d_in[5];
    const float* cl_conv_w = (const float*)d_in[6];
    const float* cl_conv_b = (const float*)d_in[7];
    const float* cl_lin_w  = (const float*)d_in[8];
    const float* cl_lin_b  = (const float*)d_in[9];
    const float* rw        = (const float*)d_in[10];
    const float* fc_w      = (const float*)d_in[11];
    const float* fc_b      = (const float*)d_in[12];
    float* out = (float*)d_out;

    const int B = in_sizes[0] / 2048;          // BATCH (2048)

    float* wsf       = (float*)d_ws;
    float* shortPart = wsf;                    // B * 2 segs * 2 outs
    float* longPart  = wsf + (size_t)B * 4;    // B * 8 segs * 2 outs

    // short: L=2048, SEG=1024 -> 2 segs/row; waves = B*2; 8 waves/block
    fused_branch_kernel<3, 2048, 1, 1024>
        <<<(B * 2) / 8, 256, 0, stream>>>(xs, cs_conv_w, cs_conv_b, cs_lin_w, shortPart);
    // long: L=8192, SEG=1024 -> 8 segs/row; waves = B*8
    fused_branch_kernel<5, 8192, 2, 1024>
        <<<(B * 8) / 8, 256, 0, stream>>>(xl, cl_conv_w, cl_conv_b, cl_lin_w, longPart);

    quantum_fc_kernel<<<(B + 127) / 128, 128, 0, stream>>>(
        shortPart, longPart, cs_lin_b, cl_lin_b, rw, fc_w, fc_b, out, B);
}